// SUFFICENT_53274774340168
// MI455X (gfx1250) — compile-verified
//
#include <hip/hip_runtime.h>
#include <hip/hip_bf16.h>

// CDNA5 / gfx1250, wave32.
// Each wave handles 16 consecutive b's: lane&15 = b column, lane>>4 = which
// half of the S=128 samples (64 samples/lane). Trilinear corner gathers are
// paired along x into b64 loads (half the VMEM requests on the divergent
// gather path). Epilogue reduces the 2 per-column partials with
// V_WMMA_F32_16X16X4_F32 (A = ones), D row M=0 -> lanes 0-15.

typedef __attribute__((ext_vector_type(2))) float v2f;
typedef __attribute__((ext_vector_type(8))) float v8f;
typedef v2f v2f_u __attribute__((aligned(4)));   // 4B-aligned float pair

#define DV 256
#define HV 256
#define WV 256
#define S_SAMP 128
#define GSCALE (256.0f / 255.0f)   // p * Wv/(Wv-1) - 0.5 mapping

__device__ __forceinline__ float fast_rcp(float x) {
    return __builtin_amdgcn_rcpf(x);
}

// 2*sigmoid(x)-1 == tanh(x/2)
__device__ __forceinline__ float sig2m1(float x) {
#if __has_builtin(__builtin_amdgcn_tanhf)
    return __builtin_amdgcn_tanhf(0.5f * x);
#else
    float e = __expf(-x);
    return (1.0f - e) * fast_rcp(1.0f + e);
#endif
}

__global__ __launch_bounds__(256) void psf_wmean_kernel(
    const float* __restrict__ vol,        // [256,256,256]
    const float* __restrict__ sampleGrid, // [B,3]
    const float* __restrict__ ax,         // [B,6]
    const float* __restrict__ bound,      // [B,2,3]
    const float* __restrict__ invCov,     // [B,3,3]
    const float* __restrict__ xyz_psf,    // [B,S,3]
    float* __restrict__ out,              // [B]
    int B)
{
    const int tid   = threadIdx.x;
    const int lane  = tid & 31;
    const int wave  = tid >> 5;
    const int bCol  = lane & 15;
    const int slot  = lane >> 4;                  // which 64-sample half
    const int b     = (blockIdx.x * 8 + wave) * 16 + bCol;
    const int bEff  = b < B ? b : (B - 1);        // keep EXEC full for WMMA

    // ---- per-b setup (duplicated across the 2 lanes of a column; L0 hits) ----
    const float* axp = ax + (size_t)bEff * 6;
    float rx = axp[0], ry = axp[1], rz = axp[2];
    float tx = axp[3], ty = axp[4], tz = axp[5];

    float theta = sqrtf(rx * rx + ry * ry + rz * rz);
    float kinv  = fast_rcp(fmaxf(theta, 1e-8f));
    float kx = rx * kinv, ky = ry * kinv, kz = rz * kinv;
    float s = __sinf(theta), c = __cosf(theta);
    float oc = 1.0f - c;
    // R = cI + (1-c) k k^T + s [k]_x
    float r00 = c + oc * kx * kx, r01 = oc * kx * ky - s * kz, r02 = oc * kx * kz + s * ky;
    float r10 = oc * ky * kx + s * kz, r11 = c + oc * ky * ky, r12 = oc * ky * kz - s * kx;
    float r20 = oc * kz * kx - s * ky, r21 = oc * kz * ky + s * kx, r22 = c + oc * kz * kz;

    const float* gp = sampleGrid + (size_t)bEff * 3;
    float gx = gp[0] + tx, gy = gp[1] + ty, gz = gp[2] + tz;
    float px = r00 * gx + r01 * gy + r02 * gz;
    float py = r10 * gx + r11 * gy + r12 * gz;
    float pz = r20 * gx + r21 * gy + r22 * gz;

    const float* bp = bound + (size_t)bEff * 6;
    float hx = (bp[3] - bp[0]) * 0.5f;
    float hy = (bp[4] - bp[1]) * 0.5f;
    float hz = (bp[5] - bp[2]) * 0.5f;

    const float* mp = invCov + (size_t)bEff * 9;
    float m00 = mp[0], m01 = mp[1], m02 = mp[2];
    float m10 = mp[3], m11 = mp[4], m12 = mp[5];
    float m20 = mp[6], m21 = mp[7], m22 = mp[8];

    const float* xp = xyz_psf + ((size_t)bEff * S_SAMP + (size_t)slot * 64) * 3;

    float vw_sum = 0.0f, w_sum = 0.0f;

#pragma unroll 8
    for (int i = 0; i < 64; ++i) {
        if ((i & 7) == 0) {
            // compile-time-guarded stream prefetch into near caches
            __builtin_prefetch(xp + (i + 32) * 3, 0, 3);
        }
        float sx = xp[i * 3 + 0];
        float sy = xp[i * 3 + 1];
        float sz = xp[i * 3 + 2];

        float bx = sig2m1(sx) * hx;
        float by = sig2m1(sy) * hy;
        float bz = sig2m1(sz) * hz;

        float ixf = fmaf(px + bx, GSCALE, -0.5f);
        float iyf = fmaf(py + by, GSCALE, -0.5f);
        float izf = fmaf(pz + bz, GSCALE, -0.5f);

        float xf = floorf(ixf), yf = floorf(iyf), zf = floorf(izf);
        float fx = ixf - xf, fy = iyf - yf, fz = izf - zf;
        int x0 = (int)xf, y0 = (int)yf, z0 = (int)zf;

        // per-axis masked lerp weights (validity factorizes per axis)
        float wx0 = (x0 >= 0 && x0 < WV)      ? (1.0f - fx) : 0.0f;
        float wx1 = (x0 >= -1 && x0 < WV - 1) ? fx          : 0.0f;
        float wy0 = (y0 >= 0 && y0 < HV)      ? (1.0f - fy) : 0.0f;
        float wy1 = (y0 >= -1 && y0 < HV - 1) ? fy          : 0.0f;
        float wz0 = (z0 >= 0 && z0 < DV)      ? (1.0f - fz) : 0.0f;
        float wz1 = (z0 >= -1 && z0 < DV - 1) ? fz          : 0.0f;

        // x-pair load at xc covers both x-corners; fold boundary cases into
        // pair weights (alpha for element .x = vol[xc], beta for .y = vol[xc+1]):
        //   interior: (wx0, wx1); x0 == -1: (wx1, 0); x0 >= WV-1: (0, wx0)
        int  xc = min(max(x0, 0), WV - 2);
        bool lo = x0 < 0;
        bool hi = x0 >= WV - 1;
        float alpha = lo ? wx1 : (hi ? 0.0f : wx0);
        float beta  = lo ? 0.0f : (hi ? wx0 : wx1);

        int yi0 = min(max(y0, 0), HV - 1), yi1 = min(max(y0 + 1, 0), HV - 1);
        int zi0 = min(max(z0, 0), DV - 1), zi1 = min(max(z0 + 1, 0), DV - 1);

        int zb0 = zi0 << 16, zb1 = zi1 << 16;   // z * 256*256
        int yb0 = yi0 << 8,  yb1 = yi1 << 8;    // y * 256

        v2f p00 = *(const v2f_u*)(vol + (zb0 + yb0 + xc));  // (z0,y0)
        v2f p01 = *(const v2f_u*)(vol + (zb0 + yb1 + xc));  // (z0,y1)
        v2f p10 = *(const v2f_u*)(vol + (zb1 + yb0 + xc));  // (z1,y0)
        v2f p11 = *(const v2f_u*)(vol + (zb1 + yb1 + xc));  // (z1,y1)

        float a00 = fmaf(beta, p00[1], alpha * p00[0]);
        float a01 = fmaf(beta, p01[1], alpha * p01[0]);
        float a10 = fmaf(beta, p10[1], alpha * p10[0]);
        float a11 = fmaf(beta, p11[1], alpha * p11[0]);
        float b0  = fmaf(a01, wy1, a00 * wy0);
        float b1  = fmaf(a11, wy1, a10 * wy0);
        float val = fmaf(b1, wz1, b0 * wz0);

        float qx = fmaf(m00, bx, fmaf(m01, by, m02 * bz));
        float qy = fmaf(m10, bx, fmaf(m11, by, m12 * bz));
        float qz = fmaf(m20, bx, fmaf(m21, by, m22 * bz));
        float q  = fmaf(bx, qx, fmaf(by, qy, bz * qz));
        float w  = __expf(-0.5f * q);

        vw_sum = fmaf(val, w, vw_sum);
        w_sum += w;
    }

    // ---- WMMA cross-lane reduction: D = ones(16x4) x partials(4x16) ----
    // Column n of B gets one partial from lane n and one from lane n+16
    // (different K-rows); unused K-rows are zero. A = all-ones, so
    // D[0,n] = total for b = base + n regardless of K-row assignment.
    v2f aOnes; aOnes[0] = 1.0f; aOnes[1] = 1.0f;
    v2f bVW;   bVW[0] = vw_sum; bVW[1] = 0.0f;
    v2f bW;    bW[0]  = w_sum;  bW[1]  = 0.0f;
    v8f zeroC = {};

    v8f dVW = __builtin_amdgcn_wmma_f32_16x16x4_f32(
        false, aOnes, false, bVW, (short)0, zeroC, false, false);
    v8f dW = __builtin_amdgcn_wmma_f32_16x16x4_f32(
        false, aOnes, false, bW, (short)0, zeroC, false, false);

    if (lane < 16 && b < B) {
        out[b] = dVW[0] * fast_rcp(dW[0]);
    }
}

extern "C" void kernel_launch(void* const* d_in, const int* in_sizes, int n_in,
                              void* d_out, int out_size, void* d_ws, size_t ws_size,
                              hipStream_t stream) {
    const float* vol        = (const float*)d_in[0];  // x [1,1,256,256,256]
    const float* sampleGrid = (const float*)d_in[1];  // [B,3]
    const float* ax         = (const float*)d_in[2];  // [B,6]
    const float* bound      = (const float*)d_in[3];  // [B,2,3]
    const float* invCov     = (const float*)d_in[4];  // [B,3,3]
    const float* xyz_psf    = (const float*)d_in[5];  // [B,128,3]
    // d_in[6] = psf_sigma, unused by the reference math

    int B = out_size;                       // 131072
    int blocks = (B + 127) / 128;           // 8 waves/block x 16 b's/wave
    psf_wmean_kernel<<<blocks, 256, 0, stream>>>(
        vol, sampleGrid, ax, bound, invCov, xyz_psf, (float*)d_out, B);
}